// Simplify_FreqFusion_87290915324089
// MI455X (gfx1250) — compile-verified
//
#include <hip/hip_runtime.h>

typedef float v2f __attribute__((ext_vector_type(2)));
typedef float v8f __attribute__((ext_vector_type(8)));

// ---------------------------------------------------------------------------
// conv1x1 as f32 WMMA GEMM: out[n,co,p] = bias[co] + sum_ci w[co,ci]*in[n,ci,p]
// Weights staged in LDS once per block; one wave computes ALL COUT rows for a
// 16-pixel tile (MT = COUT/16 accumulators), so each B fragment is loaded
// from global exactly once per K-step.  All dims compile-time so channel
// strides fold into immediate offsets of global/ds loads & stores.
// A layout (16x4 f32): lanes 0-15 hold K={0,1}, lanes 16-31 hold K={2,3};
// B mirrors with N = lane&15.
// ---------------------------------------------------------------------------
template <int CIN, int COUT, int P>
__global__ __launch_bounds__(256) void k_conv1x1_wmma(
    const float* __restrict__ in, const float* __restrict__ w,
    const float* __restrict__ bias, float* __restrict__ out,
    long long totalTiles)
{
  constexpr int MT = COUT / 16;
  __shared__ float sw[COUT * CIN];
  for (int idx = threadIdx.x; idx < COUT * CIN; idx += blockDim.x)
    sw[idx] = w[idx];
  __syncthreads();

  const int lane = threadIdx.x & 31;
  const long long tile =
      (long long)blockIdx.x * (blockDim.x >> 5) + (threadIdx.x >> 5);
  if (tile >= totalTiles) return;  // wave-uniform
  const long long p0 = tile << 4;  // global pixel over N*P (P % 16 == 0)
  const int n  = (int)(p0 / P);
  const int hw = (int)(p0 % P);
  const int half = lane >> 4;
  const int lid  = lane & 15;
  const float* __restrict__ ip = in + ((long long)n * CIN) * P + hw + lid;

  v8f acc[MT];
#pragma unroll
  for (int mt = 0; mt < MT; ++mt)
#pragma unroll
    for (int v = 0; v < 8; ++v) acc[mt][v] = bias[mt * 16 + 8 * half + v];

#pragma unroll 2
  for (int k = 0; k < CIN; k += 4) {
    const int ka = k + half * 2;
    v2f B;
    B[0] = ip[ka * P];
    B[1] = ip[(ka + 1) * P];
#pragma unroll
    for (int mt = 0; mt < MT; ++mt) {
      v2f A;
      A[0] = sw[(mt * 16 + lid) * CIN + ka];      // contiguous pair ->
      A[1] = sw[(mt * 16 + lid) * CIN + ka + 1];  // ds_load_b64 / 2addr
      acc[mt] = __builtin_amdgcn_wmma_f32_16x16x4_f32(
          false, A, false, B, (short)0, acc[mt], false, false);
    }
  }

  float* __restrict__ op = out + ((long long)n * COUT) * P + hw + lid;
#pragma unroll
  for (int mt = 0; mt < MT; ++mt)
#pragma unroll
    for (int v = 0; v < 8; ++v)
      op[(mt * 16 + 8 * half + v) * P] = acc[mt][v];  // immediate offsets
}

// ---------------------------------------------------------------------------
// conv3x3 (pad=1) as f32 WMMA GEMM with on-the-fly im2col.
// K = CIN*9 with k = ci*9 + (dy*3+dx)  (matches OIHW weight flattening).
// Weights staged in LDS; one wave computes all (padded) COUT rows for a
// 16-pixel tile (16 consecutive x in one row; W % 16 == 0).
// ---------------------------------------------------------------------------
template <int CIN, int COUT, int H, int W>
__global__ __launch_bounds__(256) void k_conv3x3_wmma(
    const float* __restrict__ in, const float* __restrict__ w,
    const float* __restrict__ bias, float* __restrict__ out,
    long long totalTiles)
{
  constexpr int K = CIN * 9;
  constexpr int MT = (COUT + 15) / 16;
  constexpr int HW = H * W;
  __shared__ float sw[COUT * K];
  for (int idx = threadIdx.x; idx < COUT * K; idx += blockDim.x)
    sw[idx] = w[idx];
  __syncthreads();

  const int lane = threadIdx.x & 31;
  const long long tile =
      (long long)blockIdx.x * (blockDim.x >> 5) + (threadIdx.x >> 5);
  if (tile >= totalTiles) return;
  const long long p0 = tile << 4;
  const int n  = (int)(p0 / HW);
  const int y  = (int)((p0 % HW) / W);
  const int x0 = (int)(p0 % W);
  const int half = lane >> 4;
  const int lid  = lane & 15;
  const float* __restrict__ ip = in + (long long)n * CIN * HW;

  v8f acc[MT];
#pragma unroll
  for (int mt = 0; mt < MT; ++mt)
#pragma unroll
    for (int v = 0; v < 8; ++v) {
      const int co = mt * 16 + 8 * half + v;
      acc[mt][v] = (co < COUT) ? bias[co] : 0.0f;
    }

  for (int k = 0; k < K; k += 4) {
    const int ka = k + half * 2;
    v2f B;
#pragma unroll
    for (int j = 0; j < 2; ++j) {
      const int kk = ka + j;
      const int ci = kk / 9;
      const int t  = kk - ci * 9;
      const int dy = t / 3 - 1;
      const int dx = t - (t / 3) * 3 - 1;
      const int yy = y + dy;
      const int xx = x0 + lid + dx;
      B[j] = (yy >= 0 && yy < H && xx >= 0 && xx < W)
                 ? ip[(ci * H + yy) * W + xx]
                 : 0.0f;
    }
#pragma unroll
    for (int mt = 0; mt < MT; ++mt) {
      const int row = mt * 16 + lid;
      v2f A;
      if (row < COUT) {
        A[0] = sw[row * K + ka];
        A[1] = sw[row * K + ka + 1];
      } else {
        A[0] = 0.0f;
        A[1] = 0.0f;
      }
      acc[mt] = __builtin_amdgcn_wmma_f32_16x16x4_f32(
          false, A, false, B, (short)0, acc[mt], false, false);
    }
  }

  float* __restrict__ op = out + (long long)n * COUT * HW + y * W + x0 + lid;
#pragma unroll
  for (int mt = 0; mt < MT; ++mt)
#pragma unroll
    for (int v = 0; v < 8; ++v) {
      const int co = mt * 16 + 8 * half + v;
      if (co < COUT) op[co * HW] = acc[mt][v];
    }
}

// ---------------------------------------------------------------------------
// kernel_normalizer: softmax over K2 channels, then renormalize by the sum
// (faithful to the reference's double normalization).
// ---------------------------------------------------------------------------
template <int K2, int HW>
__global__ __launch_bounds__(256) void k_norm_mask(
    const float* __restrict__ in, float* __restrict__ out, long long total)
{
  const long long i = (long long)blockIdx.x * blockDim.x + threadIdx.x;
  if (i >= total) return;  // total = N * HW
  const int n  = (int)(i / HW);
  const int hw = (int)(i % HW);
  const float* __restrict__ p = in + ((long long)n * K2) * HW + hw;
  float v[K2];
  float mx = -3.402823466e38f;
#pragma unroll
  for (int k = 0; k < K2; ++k) {
    v[k] = p[k * HW];
    mx = fmaxf(mx, v[k]);
  }
  float s = 0.0f;
#pragma unroll
  for (int k = 0; k < K2; ++k) {
    v[k] = __expf(v[k] - mx);
    s += v[k];
  }
  const float inv = 1.0f / s;
  float s2 = 0.0f;
#pragma unroll
  for (int k = 0; k < K2; ++k) {
    v[k] *= inv;
    s2 += v[k];
  }
  const float inv2 = 1.0f / s2;
  float* __restrict__ o = out + ((long long)n * K2) * HW + hw;
#pragma unroll
  for (int k = 0; k < K2; ++k) o[k * HW] = v[k] * inv2;
}

// ---------------------------------------------------------------------------
// CARAFE k=3, s=1 with fused residual: out = 2*feat - sum_k feat_shift * mask
// ---------------------------------------------------------------------------
template <int C, int H, int W>
__global__ __launch_bounds__(256) void k_carafe3_res(
    const float* __restrict__ feat, const float* __restrict__ mask,
    float* __restrict__ out, long long total)
{
  constexpr long long HWll = (long long)H * W;
  const long long i = (long long)blockIdx.x * blockDim.x + threadIdx.x;
  if (i >= total) return;
  const int x = (int)(i % W);
  long long t = i / W;
  const int y = (int)(t % H);
  t /= H;
  const int c = (int)(t % C);
  const int n = (int)(t / C);
  const float* __restrict__ f = feat + ((long long)n * C + c) * HWll;
  const float* __restrict__ m = mask + (long long)n * 9 * HWll + y * W + x;
  // streamed from HBM for hr_feat: hint the next row into cache
  if (y + 2 < H) __builtin_prefetch(&f[(y + 2) * W + x], 0, 3);
  float acc = 0.0f;
#pragma unroll
  for (int ki = 0; ki < 3; ++ki) {
    const int yy = y + ki - 1;
    if (yy < 0 || yy >= H) continue;
#pragma unroll
    for (int kj = 0; kj < 3; ++kj) {
      const int xx = x + kj - 1;
      if (xx < 0 || xx >= W) continue;
      acc += f[yy * W + xx] * m[(ki * 3 + kj) * HWll];
    }
  }
  out[i] = 2.0f * f[y * W + x] - acc;
}

// ---------------------------------------------------------------------------
// CARAFE k=5, s=2 (2x upsample) with optional fused addend:
//   out[n,c,yo,xo] = addend + sum_{ki,kj} feat[n,c,yo/2+ki-2,xo/2+kj-2] *
//                                         mask[n,ki*5+kj,yo,xo]
// ---------------------------------------------------------------------------
template <int C, int HLO, int WLO, bool HASADD>
__global__ __launch_bounds__(256) void k_carafe5_up2(
    const float* __restrict__ feat,    // [N,C,h,w]
    const float* __restrict__ mask,    // [N,25,2h,2w]
    const float* __restrict__ addend,  // [N,C,2h,2w] if HASADD
    float* __restrict__ out, long long total)
{
  constexpr int H = HLO * 2, W = WLO * 2;
  constexpr long long HWll = (long long)H * W;
  const long long i = (long long)blockIdx.x * blockDim.x + threadIdx.x;
  if (i >= total) return;
  const int xo = (int)(i % W);
  long long t = i / W;
  const int yo = (int)(t % H);
  t /= H;
  const int c = (int)(t % C);
  const int n = (int)(t / C);
  const int yi = yo >> 1, xi = xo >> 1;
  const float* __restrict__ f = feat + ((long long)n * C + c) * HLO * WLO;
  const float* __restrict__ m = mask + (long long)n * 25 * HWll + yo * W + xo;
  if (yi + 3 < HLO) __builtin_prefetch(&f[(yi + 3) * WLO + xi], 0, 3);
  float acc = 0.0f;
#pragma unroll
  for (int ki = 0; ki < 5; ++ki) {
    const int yy = yi + ki - 2;
    if (yy < 0 || yy >= HLO) continue;
#pragma unroll
    for (int kj = 0; kj < 5; ++kj) {
      const int xx = xi + kj - 2;
      if (xx < 0 || xx >= WLO) continue;
      acc += f[yy * WLO + xx] * m[(ki * 5 + kj) * HWll];
    }
  }
  if (HASADD) acc += addend[i];
  out[i] = acc;
}

// ---------------------------------------------------------------------------
// Host pipeline
// ---------------------------------------------------------------------------
extern "C" void kernel_launch(void* const* d_in, const int* in_sizes, int n_in,
                              void* d_out, int out_size, void* d_ws,
                              size_t ws_size, hipStream_t stream)
{
  (void)in_sizes; (void)n_in; (void)out_size; (void)ws_size;

  const float* hr_feat = (const float*)d_in[0];  // [4,256,128,128]
  const float* lr_feat = (const float*)d_in[1];  // [4,256,64,64]
  const float* hr_w    = (const float*)d_in[2];  // [64,256,1,1]
  const float* hr_b    = (const float*)d_in[3];
  const float* lr_w    = (const float*)d_in[4];  // [64,256,1,1]
  const float* lr_b    = (const float*)d_in[5];
  const float* enc_w   = (const float*)d_in[6];  // [25,64,3,3]
  const float* enc_b   = (const float*)d_in[7];
  const float* enc2_w  = (const float*)d_in[8];  // [9,64,3,3]
  const float* enc2_b  = (const float*)d_in[9];

  constexpr int N = 4, CH = 256, CC = 64;
  constexpr int H = 128, W = 128, h = 64, w = 64;
  constexpr int P_hr = H * W;   // 16384
  constexpr int P_lr = h * w;   // 4096

  // ---- workspace bump allocator (f32) ----
  float* p = (float*)d_ws;
  auto alloc = [&](long long nElems) { float* r = p; p += nElems; return r; };
  float* chf         = alloc((long long)N * CC * P_hr);
  float* clf         = alloc((long long)N * CC * P_lr);
  float* mask_hr_hr  = alloc((long long)N * 9 * P_hr);
  float* mask_hr_in  = alloc((long long)N * 9 * P_hr);
  float* chf2        = alloc((long long)N * CC * P_hr);
  float* mask_lr_hr  = alloc((long long)N * 25 * P_hr);
  float* mask_lr_i1  = alloc((long long)N * 25 * P_hr);
  float* mask_lr_ll  = alloc((long long)N * 25 * P_lr);
  float* mask_lr     = alloc((long long)N * 25 * P_hr);
  float* enc2clf     = alloc((long long)N * 9 * P_lr);
  float* mask_hr     = alloc((long long)N * 9 * P_hr);
  float* mask_hr_out = alloc((long long)N * 9 * P_hr);

  // ---- outputs, concatenated flat in return order ----
  float* o_mask_lr = (float*)d_out;                         // [4,25,128,128]
  float* o_hr = o_mask_lr + (long long)N * 25 * P_hr;       // [4,256,128,128]
  float* o_lr = o_hr + (long long)N * CH * P_hr;            // [4,256,128,128]

  const int BLK = 256;  // 8 wave32 waves per block
  auto wgrid = [](long long tiles) { return (unsigned)((tiles + 7) / 8); };
  auto egrid = [](long long n) { return (unsigned)((n + 255) / 256); };

  // 1) chf = conv1x1(hr_feat, hr_w) + hr_b     (4096 pixel tiles)
  {
    long long tiles = (long long)N * P_hr / 16;
    k_conv1x1_wmma<256, 64, P_hr><<<wgrid(tiles), BLK, 0, stream>>>(
        hr_feat, hr_w, hr_b, chf, tiles);
  }
  // 2) clf = conv1x1(lr_feat, lr_w) + lr_b     (1024 pixel tiles)
  {
    long long tiles = (long long)N * P_lr / 16;
    k_conv1x1_wmma<256, 64, P_lr><<<wgrid(tiles), BLK, 0, stream>>>(
        lr_feat, lr_w, lr_b, clf, tiles);
  }
  // 3) mask_hr_hr = conv3x3(chf, enc2_w)
  {
    long long tiles = (long long)N * P_hr / 16;
    k_conv3x3_wmma<64, 9, H, W><<<wgrid(tiles), BLK, 0, stream>>>(
        chf, enc2_w, enc2_b, mask_hr_hr, tiles);
  }
  // 4) mask_hr_init = normalize_k3(mask_hr_hr)
  {
    long long tot = (long long)N * P_hr;
    k_norm_mask<9, P_hr><<<egrid(tot), BLK, 0, stream>>>(mask_hr_hr,
                                                         mask_hr_in, tot);
  }
  // 5) chf2 = 2*chf - carafe3(chf, mask_hr_init)
  {
    long long tot = (long long)N * CC * P_hr;
    k_carafe3_res<CC, H, W><<<egrid(tot), BLK, 0, stream>>>(chf, mask_hr_in,
                                                            chf2, tot);
  }
  // 6) mask_lr_hr = conv3x3(chf2, enc_w)
  {
    long long tiles = (long long)N * P_hr / 16;
    k_conv3x3_wmma<64, 25, H, W><<<wgrid(tiles), BLK, 0, stream>>>(
        chf2, enc_w, enc_b, mask_lr_hr, tiles);
  }
  // 7) mask_lr_init = normalize_k5(mask_lr_hr)
  {
    long long tot = (long long)N * P_hr;
    k_norm_mask<25, P_hr><<<egrid(tot), BLK, 0, stream>>>(mask_lr_hr,
                                                          mask_lr_i1, tot);
  }
  // 8) mask_lr_lr_lr = conv3x3(clf, enc_w)
  {
    long long tiles = (long long)N * P_lr / 16;
    k_conv3x3_wmma<64, 25, h, w><<<wgrid(tiles), BLK, 0, stream>>>(
        clf, enc_w, enc_b, mask_lr_ll, tiles);
  }
  // 9) mask_lr = mask_lr_hr + carafe5_up2(mask_lr_lr_lr, mask_lr_init)
  {
    long long tot = (long long)N * 25 * P_hr;
    k_carafe5_up2<25, h, w, true><<<egrid(tot), BLK, 0, stream>>>(
        mask_lr_ll, mask_lr_i1, mask_lr_hr, mask_lr, tot);
  }
  // 10) mask_lr_out = normalize_k5(mask_lr)  -> output 0 (also reused below)
  {
    long long tot = (long long)N * P_hr;
    k_norm_mask<25, P_hr><<<egrid(tot), BLK, 0, stream>>>(mask_lr, o_mask_lr,
                                                          tot);
  }
  // 11) enc2clf = conv3x3(clf, enc2_w)
  {
    long long tiles = (long long)N * P_lr / 16;
    k_conv3x3_wmma<64, 9, h, w><<<wgrid(tiles), BLK, 0, stream>>>(
        clf, enc2_w, enc2_b, enc2clf, tiles);
  }
  // 12) mask_hr = mask_hr_hr + carafe5_up2(enc2clf, mask_lr_out)
  {
    long long tot = (long long)N * 9 * P_hr;
    k_carafe5_up2<9, h, w, true><<<egrid(tot), BLK, 0, stream>>>(
        enc2clf, o_mask_lr, mask_hr_hr, mask_hr, tot);
  }
  // 13) mask_hr_out = normalize_k3(mask_hr)
  {
    long long tot = (long long)N * P_hr;
    k_norm_mask<9, P_hr><<<egrid(tot), BLK, 0, stream>>>(mask_hr, mask_hr_out,
                                                         tot);
  }
  // 14) hr_out = 2*hr_feat - carafe3(hr_feat, mask_hr_out)  -> output 1
  {
    long long tot = (long long)N * CH * P_hr;
    k_carafe3_res<CH, H, W><<<egrid(tot), BLK, 0, stream>>>(
        hr_feat, mask_hr_out, o_hr, tot);
  }
  // 15) lr_out = carafe5_up2(lr_feat, mask_lr_out)  -> output 2
  {
    long long tot = (long long)N * CH * P_hr;
    k_carafe5_up2<CH, h, w, false><<<egrid(tot), BLK, 0, stream>>>(
        lr_feat, o_mask_lr, nullptr, o_lr, tot);
  }
}